// permuter_54219667144894
// MI455X (gfx1250) — compile-verified
//
#include <hip/hip_runtime.h>

#define BATCH 32
#define N 1024

typedef __attribute__((ext_vector_type(2))) float v2f;
typedef __attribute__((ext_vector_type(8))) float v8f;

// ---------------------------------------------------------------------------
// Stage 1: per-row soft ranks. One 1024-thread block per batch row.
//   - LDS bitonic sort (descending) of theta = x (EPS = 1) with index payload
//   - y = s - [n, n-1, ..., 1]
//   - PAV non-increasing isotonic regression (thread 0, O(n) amortized)
//   - parallel binary-search expansion; scatter ranks = s - sol to original pos
// ---------------------------------------------------------------------------
__global__ __launch_bounds__(1024)
void softrank_rank_kernel(const float* __restrict__ x, float* __restrict__ ranks) {
    __shared__ float vals[N];    // sorted values s
    __shared__ int   idxA[N];    // original positions (perm)
    __shared__ float yv[N];      // s - w
    __shared__ float bSum[N];    // PAV block stack
    __shared__ float bCnt[N];
    __shared__ int   bStart[N];
    __shared__ float bAvg[N];
    __shared__ int   nbS;

    const int tid = threadIdx.x;
    const int row = blockIdx.x;

    vals[tid] = x[row * N + tid];
    idxA[tid] = tid;
    __syncthreads();

    // Bitonic sort, overall descending (final merge k==N has (tid & k)==0 for all tid)
    for (int k = 2; k <= N; k <<= 1) {
        for (int j = k >> 1; j > 0; j >>= 1) {
            int ixj = tid ^ j;
            if (ixj > tid) {
                bool desc = ((tid & k) == 0);
                float a = vals[tid], b = vals[ixj];
                bool doSwap = desc ? (a < b) : (a > b);
                if (doSwap) {
                    vals[tid] = b; vals[ixj] = a;
                    int ia = idxA[tid]; idxA[tid] = idxA[ixj]; idxA[ixj] = ia;
                }
            }
            __syncthreads();
        }
    }

    // y[p] = s[p] - w[p],  w = [N, N-1, ..., 1]
    yv[tid] = vals[tid] - (float)(N - tid);
    __syncthreads();

    // PAV: non-increasing isotonic regression of yv (merge while prev avg <= new avg)
    if (tid == 0) {
        int nb = 0;
        float topAvg = 0.0f;
        for (int i = 0; i < N; ++i) {
            float s = yv[i];
            float c = 1.0f;
            int   st = i;
            float a = s;
            while (nb > 0 && topAvg <= a) {
                nb--;
                s += bSum[nb]; c += bCnt[nb]; st = bStart[nb];
                a = s / c;
                if (nb > 0) topAvg = bAvg[nb - 1];
            }
            bSum[nb] = s; bCnt[nb] = c; bStart[nb] = st; bAvg[nb] = a;
            topAvg = a;
            nb++;
        }
        nbS = nb;
    }
    __syncthreads();

    // Find block containing sorted position tid: largest bi with bStart[bi] <= tid
    int lo = 0, hi = nbS - 1;
    while (lo < hi) {
        int mid = (lo + hi + 1) >> 1;
        if (bStart[mid] <= tid) lo = mid; else hi = mid - 1;
    }
    float sol    = bAvg[lo];
    float primal = vals[tid] - sol;          // ranks in sorted order
    ranks[row * N + idxA[tid]] = primal;     // scatter back to original order
}

// ---------------------------------------------------------------------------
// Stage 2: out[b,j,i] = relu(1 - |ranks[b,j] - (i+1)|), one 16x16 tile / wave.
// WMMA computes the rank-1 difference matrix D[m,n] = r_m - t_n:
//   A[m,0] = r_m (other K slots 0)  [A layout documented: lanes 0-15 K=0,1;
//                                    lanes 16-31 K=2,3]
//   B      = all ones                -> result independent of B lane layout
//   C[m,n] = -t_n                    [C/D layout: VGPR v, lane l -> M=v+8*(l>=16),
//                                                                 N=l&15]
// ---------------------------------------------------------------------------
__global__ __launch_bounds__(256)
void softrank_out_kernel(const float* __restrict__ ranks, float* __restrict__ out) {
    const int lane = threadIdx.x & 31;
    const int wave = threadIdx.x >> 5;
    const int tile = blockIdx.x * 8 + wave;        // 131072 tiles total
    const int bI   = tile >> 12;                   // tile / 4096
    const int rem  = tile & 4095;
    const int jt   = rem >> 6;                     // J-tile (0..63)
    const int it   = rem & 63;                     // I-tile (0..63)

    // All 32 lanes load (no EXEC divergence); lanes 16-31 get masked to 0 via select.
    float r = ranks[bI * N + jt * 16 + (lane & 15)];

    v2f a;
    a.x = (lane < 16) ? r : 0.0f;   // K=0 slot (lanes 0-15); K=2 slot (lanes 16-31) = 0
    a.y = 0.0f;                     // K=1 / K=3 slots = 0

    v2f bb;
    bb.x = 1.0f;                    // all-ones B: D[m,n] = sum_k A[m,k] + C[m,n]
    bb.y = 1.0f;

    float tcol = (float)(it * 16 + (lane & 15) + 1);   // t_n = i + 1
    v8f c;
#pragma unroll
    for (int v = 0; v < 8; ++v) c[v] = -tcol;

    // D[m,n] = r_m - t_n
    v8f d = __builtin_amdgcn_wmma_f32_16x16x4_f32(
        /*neg_a=*/false, a, /*neg_b=*/false, bb,
        /*c_mod=*/(short)0, c, /*reuse_a=*/false, /*reuse_b=*/false);

    const int mBase = (lane >> 4) << 3;   // lanes 16-31 hold rows M = v + 8
    size_t rowBase = ((size_t)bI * N + (size_t)(jt * 16)) * N
                   + (size_t)(it * 16 + (lane & 15));
#pragma unroll
    for (int v = 0; v < 8; ++v) {
        float val = fmaxf(0.0f, 1.0f - fabsf(d[v]));
        out[rowBase + (size_t)(v + mBase) * N] = val;   // two dense 64B segments / store
    }
}

extern "C" void kernel_launch(void* const* d_in, const int* in_sizes, int n_in,
                              void* d_out, int out_size, void* d_ws, size_t ws_size,
                              hipStream_t stream) {
    const float* x     = (const float*)d_in[0];
    float*       out   = (float*)d_out;
    float*       ranks = (float*)d_ws;    // 32*1024 floats = 128 KB scratch

    softrank_rank_kernel<<<BATCH, N, 0, stream>>>(x, ranks);

    const int tiles  = BATCH * (N / 16) * (N / 16);   // 131072
    const int blocks = tiles / 8;                      // 8 waves (tiles) per block
    softrank_out_kernel<<<blocks, 256, 0, stream>>>(ranks, out);
}